// FineGrainedPromptTuning_11957188952492
// MI455X (gfx1250) — compile-verified
//
#include <hip/hip_runtime.h>

// Problem constants from the reference: L=12, B=16, S=577, D=1024, H=16
#define L_DIM 12
#define B_DIM 16
#define S_DIM 577
#define D_DIM 1024
#define H_DIM 16
#define N_TOK 576           // S - 1 spatial tokens
#define K_SEL 115           // int(576 * 0.2)

typedef __attribute__((ext_vector_type(2))) float v2f;
typedef __attribute__((ext_vector_type(8))) float v8f;

// ---------------------------------------------------------------------------
// Kernel 1: per-batch head-mean scores via V_WMMA_F32_16X16X4_F32, then exact
// stable top-K (descending value, ties -> lower index) via rank counting.
// One block per batch, 512 threads = 16 wave32s.
// ---------------------------------------------------------------------------
__global__ __launch_bounds__(512)
void fgpt_score_topk(const float* __restrict__ amap, int* __restrict__ topk)
{
    __shared__ float sc[N_TOK];

    const int b    = blockIdx.x;
    const int tid  = threadIdx.x;
    const int lane = tid & 31;
    const int wid  = tid >> 5;
    const int m    = lane & 15;   // row of A (= local token index within j-block)
    const int half = lane >> 4;   // selects K=0,1 (half 0) vs K=2,3 (half 1) of A

    // amap layout (B, H, S, S): element [b, h, 0, 1 + j]
    const size_t hstride = (size_t)S_DIM * S_DIM;
    const float* abase   = amap + (size_t)b * H_DIM * hstride + 1;

    // B matrix = ones * (1/16): uniform, so its lane layout is irrelevant.
    v2f bm; bm[0] = 0.0625f; bm[1] = 0.0625f;

    // 36 j-blocks of 16 tokens; wave w handles jb = w, w+16, (w+32)
    for (int jb = wid; jb < N_TOK / 16; jb += 16) {
        const int j = jb * 16 + m;
        v8f c = {};
        #pragma unroll
        for (int g = 0; g < 4; ++g) {
            // A(m, k) = amap[b, 4g + k, 0, 1 + j]; VGPR0 holds K=2*half, VGPR1 K=2*half+1
            const int h0 = g * 4 + half * 2;
            v2f a;
            a[0] = abase[(size_t)(h0)     * hstride + j];
            a[1] = abase[(size_t)(h0 + 1) * hstride + j];
            // D(m, n) = sum_k A(m,k)/16 + C  (all 16 columns identical)
            c = __builtin_amdgcn_wmma_f32_16x16x4_f32(false, a, false, bm,
                                                      (short)0, c, false, false);
        }
        // C/D layout: lanes 0-15 VGPR r = row M=r; lanes 16-31 VGPR r = row M=8+r.
        // Lane 0 writes tokens jb*16+0..7, lane 16 writes jb*16+8..15.
        if (m == 0) {
            const int base = jb * 16 + half * 8;
            #pragma unroll
            for (int r = 0; r < 8; ++r) sc[base + r] = c[r];
        }
    }
    __syncthreads();

    // Exact lax.top_k: rank = #(greater) + #(equal with smaller index).
    for (int i = tid; i < N_TOK; i += 512) {
        const float si = sc[i];
        int cnt = 0;
        for (int jj = 0; jj < N_TOK; ++jj) {
            const float sj = sc[jj];
            cnt += (sj > si) || ((sj == si) && (jj < i));
        }
        if (cnt < K_SEL) topk[b * K_SEL + cnt] = i;
    }
}

// ---------------------------------------------------------------------------
// Kernel 2: bandwidth-bound gather. One block per (l, b, k) row; each thread
// moves one float4 (b128) of the key row and one of the value row. Rows are
// 4 KB contiguous -> fully coalesced; direct global->global is the roofline-
// optimal path (no reuse, so no LDS staging).
// ---------------------------------------------------------------------------
__global__ __launch_bounds__(256)
void fgpt_gather(const float4* __restrict__ keys,
                 const float4* __restrict__ vals,
                 const int*    __restrict__ topk,
                 float4*       __restrict__ out)
{
    const int row = blockIdx.x;                 // 0 .. L*B*K-1
    const int k   = row % K_SEL;
    const int b   = (row / K_SEL) % B_DIM;
    const int l   = row / (K_SEL * B_DIM);
    const int tid = threadIdx.x;                // D/4 = 256 float4 per row

    const int    s      = 1 + topk[b * K_SEL + k];          // spatial index in S
    const size_t srcRow = (((size_t)l * B_DIM + b) * S_DIM + s) * (D_DIM / 4);
    const size_t dstRow = (((size_t)l * B_DIM + b) * K_SEL + k) * (D_DIM / 4);
    const size_t valOut = (size_t)L_DIM * B_DIM * K_SEL * (D_DIM / 4);

    out[dstRow + tid]          = keys[srcRow + tid];
    out[valOut + dstRow + tid] = vals[srcRow + tid];
}

// ---------------------------------------------------------------------------
extern "C" void kernel_launch(void* const* d_in, const int* in_sizes, int n_in,
                              void* d_out, int out_size, void* d_ws, size_t ws_size,
                              hipStream_t stream)
{
    const float* key_states  = (const float*)d_in[0];   // (L, B, S, D) f32
    const float* value_states= (const float*)d_in[1];   // (L, B, S, D) f32
    const float* attention   = (const float*)d_in[2];   // (B, H, S, S) f32

    int* topk = (int*)d_ws;                             // B*K ints = 7360 bytes

    fgpt_score_topk<<<B_DIM, 512, 0, stream>>>(attention, topk);

    const int nrows = L_DIM * B_DIM * K_SEL;            // 22080
    fgpt_gather<<<nrows, 256, 0, stream>>>(
        (const float4*)key_states, (const float4*)value_states,
        topk, (float4*)d_out);
}